// EncodingNet_84902913507381
// MI455X (gfx1250) — compile-verified
//
#include <hip/hip_runtime.h>
#include <hip/hip_bf16.h>

// Problem constants (fixed by the reference).
namespace {
constexpr int N    = 2048;   // nodes
constexpr int C    = 2;      // channels
constexpr int T    = 3;      // edge types
constexpr int E    = 65536;  // edges per type (power of two)
constexpr int NE   = T * E;  // 196608 edges total
constexpr int WIN  = 256;
constexpr int WOUT = 64;
constexpr int HID  = 16;     // gcn hidden
constexpr int NCLS = 16;
constexpr int NTGT = 512;
}

typedef __attribute__((ext_vector_type(2))) float v2f;
typedef __attribute__((ext_vector_type(8))) float v8f;

// ---------------------------------------------------------------------------
// Dense fp32 GEMM tile kernel: C[M,Nd] = A[M,K] @ B[K,Nd] (+bias), row-major.
// One wave32 computes one 16x16 tile via V_WMMA_F32_16X16X4_F32.
// Requires M%16==0, Nd%16==0, K%4==0 (all call sites satisfy this).
// Fragment layout per ISA 7.12.2:
//   A 16x4 f32: lanes 0-15 M=lane, VGPR0/1 = K0/K1; lanes 16-31 = K2/K3.
//   B 4x16 f32: VGPR v, lane half h -> row K = v + 2h, col N = lane&15.
//   C/D 16x16 f32: VGPR v -> row M = v + 8*(lane>>4), col N = lane&15.
// ---------------------------------------------------------------------------
__global__ void gemm16_f32_wmma(const float* __restrict__ A,
                                const float* __restrict__ B,
                                float* __restrict__ Cm,
                                int M, int Nd, int K,
                                const float* __restrict__ bias) {
  const int lane   = threadIdx.x & 31;
  const int wave   = threadIdx.x >> 5;
  const int tilesN = Nd >> 4;
  const int tile   = blockIdx.x * (blockDim.x >> 5) + wave;
  if (tile >= (M >> 4) * tilesN) return;   // wave-uniform: EXEC stays all-ones
  const int tm   = tile / tilesN;
  const int tn   = tile - tm * tilesN;
  const int m0   = tm << 4;
  const int n0   = tn << 4;
  const int half = lane >> 4;   // selects K-pair (A) / K rows (B)
  const int l15  = lane & 15;   // M row for A, N col for B/C

  const float* arow = A + (long)(m0 + l15) * K;
  const float* bcol = B + n0 + l15;

  v8f acc = {0.f, 0.f, 0.f, 0.f, 0.f, 0.f, 0.f, 0.f};
  for (int k0 = 0; k0 < K; k0 += 4) {
    const int ka = k0 + 2 * half;
    v2f a, b;
    a.x = arow[ka];
    a.y = arow[ka + 1];
    b.x = bcol[(long)ka * Nd];
    b.y = bcol[(long)(ka + 1) * Nd];
    acc = __builtin_amdgcn_wmma_f32_16x16x4_f32(false, a, false, b,
                                                (short)0, acc, false, false);
  }
  const float bv = bias ? bias[n0 + l15] : 0.0f;
#pragma unroll
  for (int v = 0; v < 8; ++v) {
    Cm[(long)(m0 + v + 8 * half) * Nd + n0 + l15] = acc[v] + bv;
  }
}

// ---------------------------------------------------------------------------
// Edge scatter SpMM: out[dst,:] += w_e * in[src,:] over all T*E edges.
// w_e = edge_value[t,e] * filt[lc, t];  dst = edge_index[t,0,e], src = [t,1,e].
// in == nullptr means multiply by ones-vector (degree SpMV, F==1).
// ---------------------------------------------------------------------------
template <int F>
__global__ void spmm_scatter(const int* __restrict__ ei,    // [T,2,E] int32
                             const float* __restrict__ ev,  // [T,E]
                             const float* __restrict__ filt, int lc,
                             const float* __restrict__ in,
                             float* __restrict__ out) {
  const long gid = (long)blockIdx.x * blockDim.x + threadIdx.x;
  if (gid >= (long)NE * F) return;
  const int e  = (int)(gid / F);
  const int f  = (int)(gid - (long)e * F);
  const int t  = e >> 16;        // e / E, E = 65536
  const int ee = e & (E - 1);
  const int dst = ei[(t * 2) * E + ee];       // row index
  const int src = ei[(t * 2 + 1) * E + ee];   // col index
  const float w = ev[e] * filt[lc * T + t];
  const float x = (in == nullptr) ? 1.0f : in[(long)src * F + f];
  atomicAdd(&out[(long)dst * F + f], w * x);
}

// softmax over T=3 edge-type logits for each of L*C=4 (layer,channel) rows.
__global__ void softmax_filt(const float* __restrict__ conv_w,
                             float* __restrict__ filt) {
  const int i = threadIdx.x;
  if (i < 4) {
    float a = conv_w[i * 3], b = conv_w[i * 3 + 1], c = conv_w[i * 3 + 2];
    float m  = fmaxf(a, fmaxf(b, c));
    float ea = expf(a - m), eb = expf(b - m), ec = expf(c - m);
    float s  = ea + eb + ec;
    filt[i * 3]     = ea / s;
    filt[i * 3 + 1] = eb / s;
    filt[i * 3 + 2] = ec / s;
  }
}

// Hc = mean_c relu(0.5*X_ + 0.5*H2)   (BETA = 0.5, C = 2)
__global__ void hc_kernel(const float* __restrict__ X_,
                          const float* __restrict__ H2,
                          float* __restrict__ Hc) {
  const int i = blockIdx.x * blockDim.x + threadIdx.x;
  if (i >= N * WOUT) return;
  const float a = fmaxf(0.5f * X_[i] + 0.5f * H2[i], 0.0f);
  const float b = fmaxf(0.5f * X_[N * WOUT + i] + 0.5f * H2[N * WOUT + i], 0.0f);
  Hc[i] = 0.5f * (a + b);
}

// dinv = rsqrt(deg) with deg = (S1@(S0@1)) summed over channels + 1 (self loop)
__global__ void dinv_kernel(const float* __restrict__ t2,
                            float* __restrict__ dinv) {
  const int i = blockIdx.x * blockDim.x + threadIdx.x;
  if (i >= N) return;
  const float d = t2[i] + 1.0f;
  dinv[i] = (d > 0.0f) ? rsqrtf(d) : 0.0f;
}

template <int F>
__global__ void rowscale(const float* __restrict__ dinv, float* __restrict__ Y) {
  const int i = blockIdx.x * blockDim.x + threadIdx.x;
  if (i >= N * F) return;
  Y[i] *= dinv[i / F];
}

// out = [relu]( dinv .* (Z + R) + bias )   -- completes An@Y + bias
template <int F, bool RELU>
__global__ void an_combine(const float* __restrict__ dinv,
                           const float* __restrict__ Z,
                           const float* __restrict__ R,
                           const float* __restrict__ bias,
                           float* __restrict__ out) {
  const int i = blockIdx.x * blockDim.x + threadIdx.x;
  if (i >= N * F) return;
  const int row = i / F;
  const int f   = i - row * F;
  float v = dinv[row] * (Z[i] + R[i]) + bias[f];
  out[i] = RELU ? fmaxf(v, 0.0f) : v;
}

// In-place log_softmax over each 64-wide row; one wave32 per row.
__global__ void log_softmax64(float* __restrict__ h) {
  const int lane = threadIdx.x & 31;
  const int row  = blockIdx.x * (blockDim.x >> 5) + (threadIdx.x >> 5);
  if (row >= N) return;
  float v0 = h[row * 64 + lane];
  float v1 = h[row * 64 + 32 + lane];
  float m = fmaxf(v0, v1);
  for (int off = 16; off; off >>= 1) m = fmaxf(m, __shfl_xor(m, off, 32));
  float s = expf(v0 - m) + expf(v1 - m);
  for (int off = 16; off; off >>= 1) s += __shfl_xor(s, off, 32);
  const float ls = logf(s);
  h[row * 64 + lane]      = v0 - m - ls;
  h[row * 64 + 32 + lane] = v1 - m - ls;
}

__global__ void gather_rows(const float* __restrict__ h,
                            const int* __restrict__ tgt,
                            float* __restrict__ G) {
  const int i = blockIdx.x * blockDim.x + threadIdx.x;
  if (i >= NTGT * WOUT) return;
  const int r = i / WOUT;
  const int f = i - r * WOUT;
  G[i] = h[(long)tgt[r] * WOUT + f];
}

// ---------------------------------------------------------------------------
extern "C" void kernel_launch(void* const* d_in, const int* in_sizes, int n_in,
                              void* d_out, int out_size, void* d_ws, size_t ws_size,
                              hipStream_t stream) {
  const float* X      = (const float*)d_in[0];   // [N, WIN]
  const float* ev     = (const float*)d_in[1];   // [T, E]
  const float* conv_w = (const float*)d_in[2];   // [L, C, T]
  const float* Ws     = (const float*)d_in[3];   // [C, WIN, WOUT]
  const float* gw1    = (const float*)d_in[4];   // [WOUT, HID]
  const float* gb1    = (const float*)d_in[5];   // [HID]
  const float* gw2    = (const float*)d_in[6];   // [HID, WOUT]
  const float* gb2    = (const float*)d_in[7];   // [WOUT]
  const float* lw     = (const float*)d_in[8];   // [WOUT, NCLS]
  const float* lb     = (const float*)d_in[9];   // [NCLS]
  const int*   ei     = (const int*)d_in[10];    // [T, 2, E]
  const int*   tgt    = (const int*)d_in[11];    // [NTGT]
  float* y = (float*)d_out;                      // [NTGT, NCLS]

  // Workspace carve-up (~5.4 MB of fp32).
  float* w    = (float*)d_ws;
  float* filt = w; w += 16;
  float* X_   = w; w += (long)C * N * WOUT;   // per-channel projection
  float* Htmp = w; w += (long)N * WOUT;       // S0 @ X_  (also reused as T64)
  float* H2   = w; w += (long)C * N * WOUT;   // S1 @ S0 @ X_ per channel
  float* Hc   = w; w += (long)N * WOUT;
  float* t1   = w; w += N;
  float* t2   = w; w += N;
  float* dinv = w; w += N;
  float* P    = w; w += (long)N * HID;        // Hc@gw1, then Z (row-scaled)
  float* T16  = w; w += (long)N * HID;
  float* R16  = w; w += (long)N * HID;
  float* h1   = w; w += (long)N * HID;
  float* Q    = w; w += (long)N * WOUT;       // h1@gw2, then Z2
  float* R64  = w; w += (long)N * WOUT;
  float* h2   = w; w += (long)N * WOUT;
  float* G    = w; w += (long)NTGT * WOUT;
  float* T64  = Htmp;                          // H stage done before An stage

  auto gemm = [&](const float* A, const float* B, float* Cm,
                  int M, int Nd, int K, const float* bias) {
    const int tiles  = (M / 16) * (Nd / 16);
    const int blocks = (tiles + 3) / 4;                    // 4 waves / block
    gemm16_f32_wmma<<<blocks, 128, 0, stream>>>(A, B, Cm, M, Nd, K, bias);
  };

  const int g64 = (NE * 64 + 255) / 256;
  const int g16 = (NE * 16 + 255) / 256;
  const int g1  = (NE + 255) / 256;
  const int gN64 = (N * WOUT + 255) / 256;
  const int gN16 = (N * HID + 255) / 256;

  // 1) softmax over edge-type mixing weights
  softmax_filt<<<1, 32, 0, stream>>>(conv_w, filt);

  // 2) per-channel input projection X_ = X @ Ws[c]  (fp32 WMMA)
  for (int c = 0; c < C; ++c)
    gemm(X, Ws + (long)c * WIN * WOUT, X_ + (long)c * N * WOUT, N, WOUT, WIN, nullptr);

  // 3) H = S1_c @ S0_c @ X_c   (sparse edge scatter)
  for (int c = 0; c < C; ++c) {
    hipMemsetAsync(Htmp, 0, (size_t)N * WOUT * 4, stream);
    spmm_scatter<64><<<g64, 256, 0, stream>>>(ei, ev, filt, 0 * C + c,
                                              X_ + (long)c * N * WOUT, Htmp);
    hipMemsetAsync(H2 + (long)c * N * WOUT, 0, (size_t)N * WOUT * 4, stream);
    spmm_scatter<64><<<g64, 256, 0, stream>>>(ei, ev, filt, 1 * C + c,
                                              Htmp, H2 + (long)c * N * WOUT);
  }

  // 4) channel aggregation
  hc_kernel<<<gN64, 256, 0, stream>>>(X_, H2, Hc);

  // 5) degrees of A_comb = sum_c S1_c@S0_c + I via SpMV with ones
  hipMemsetAsync(t2, 0, (size_t)N * 4, stream);
  for (int c = 0; c < C; ++c) {
    hipMemsetAsync(t1, 0, (size_t)N * 4, stream);
    spmm_scatter<1><<<g1, 256, 0, stream>>>(ei, ev, filt, 0 * C + c, nullptr, t1);
    spmm_scatter<1><<<g1, 256, 0, stream>>>(ei, ev, filt, 1 * C + c, t1, t2);
  }
  dinv_kernel<<<(N + 255) / 256, 256, 0, stream>>>(t2, dinv);

  // 6) GCN layer 1: h1 = relu(An @ (Hc@gw1) + gb1), An@Y done sparsely
  gemm(Hc, gw1, P, N, HID, WOUT, nullptr);
  rowscale<HID><<<gN16, 256, 0, stream>>>(dinv, P);          // Z = dinv .* P
  hipMemsetAsync(R16, 0, (size_t)N * HID * 4, stream);
  for (int c = 0; c < C; ++c) {
    hipMemsetAsync(T16, 0, (size_t)N * HID * 4, stream);
    spmm_scatter<16><<<g16, 256, 0, stream>>>(ei, ev, filt, 0 * C + c, P, T16);
    spmm_scatter<16><<<g16, 256, 0, stream>>>(ei, ev, filt, 1 * C + c, T16, R16);
  }
  an_combine<HID, true><<<gN16, 256, 0, stream>>>(dinv, P, R16, gb1, h1);

  // 7) GCN layer 2: h2 = An @ (h1@gw2) + gb2
  gemm(h1, gw2, Q, N, WOUT, HID, nullptr);
  rowscale<WOUT><<<gN64, 256, 0, stream>>>(dinv, Q);
  hipMemsetAsync(R64, 0, (size_t)N * WOUT * 4, stream);
  for (int c = 0; c < C; ++c) {
    hipMemsetAsync(T64, 0, (size_t)N * WOUT * 4, stream);
    spmm_scatter<64><<<g64, 256, 0, stream>>>(ei, ev, filt, 0 * C + c, Q, T64);
    spmm_scatter<64><<<g64, 256, 0, stream>>>(ei, ev, filt, 1 * C + c, T64, R64);
  }
  an_combine<WOUT, false><<<gN64, 256, 0, stream>>>(dinv, Q, R64, gb2, h2);

  // 8) log_softmax over features, gather targets, classification head
  log_softmax64<<<N / 4, 128, 0, stream>>>(h2);
  gather_rows<<<(NTGT * WOUT + 255) / 256, 256, 0, stream>>>(h2, tgt, G);
  gemm(G, lw, y, NTGT, NCLS, WOUT, lb);
}